// Render_53412213293435
// MI455X (gfx1250) — compile-verified
//
#include <hip/hip_runtime.h>
#include <cmath>

typedef __attribute__((ext_vector_type(2))) float v2f;
typedef __attribute__((ext_vector_type(8))) float v8f;
typedef __attribute__((ext_vector_type(4))) unsigned int u32x4;
typedef __attribute__((ext_vector_type(4))) int i32x4;
typedef __attribute__((ext_vector_type(8))) int i32x8;

#define EPS_W 1e-9f

// ---------------------------------------------------------------------------
// Kernel 1: zinit = min over all vertices of z (initial depth buffer value)
// ---------------------------------------------------------------------------
__global__ void zinit_kernel(const float* __restrict__ verts, int nvert,
                             float* __restrict__ zinit) {
    __shared__ float red[256];
    float m = 3.402823466e38f;
    for (int i = threadIdx.x; i < nvert; i += blockDim.x)
        m = fminf(m, verts[3 * i + 2]);
    red[threadIdx.x] = m;
    __syncthreads();
    for (int s = 128; s > 0; s >>= 1) {
        if (threadIdx.x < s)
            red[threadIdx.x] = fminf(red[threadIdx.x], red[threadIdx.x + s]);
        __syncthreads();
    }
    if (threadIdx.x == 0) *zinit = red[0];
}

// ---------------------------------------------------------------------------
// Kernel 2: per-triangle affine coefficients, written directly in WMMA
// A-matrix (16x4 f32) lane layout, 4 triangles per 256B group:
//   row m = 4*(t%4)+f, f in {pAB,pCB,pCA,z};
//   lane m (0..15): slot0=K0(a) slot1=K1(b); lane m+16: slot0=K2(c) slot1=K3(0)
// Invalid triangles (backfacing/degenerate: w < EPS) -> all-zero coefficients,
// so the strict (>0) edge tests can never pass.  UV planes: 8 floats per tri.
// ---------------------------------------------------------------------------
__global__ void coef_kernel(const float* __restrict__ verts,
                            const int* __restrict__ faces,
                            const float* __restrict__ uv,
                            const int* __restrict__ uvfaces,
                            int ntri, int ntri_pad,
                            float* __restrict__ acoef,
                            float* __restrict__ uvcoef) {
    int t = blockIdx.x * blockDim.x + threadIdx.x;
    if (t >= ntri_pad) return;

    float aAB = 0.f, bAB = 0.f, cAB = 0.f;
    float aCB = 0.f, bCB = 0.f, cCB = 0.f;
    float aCA = 0.f, bCA = 0.f, cCA = 0.f;
    float az = 0.f, bz = 0.f, cz = 0.f;
    float au = 0.f, bu = 0.f, cu = 0.f;
    float av = 0.f, bv = 0.f, cv = 0.f;

    if (t < ntri) {
        int i0 = faces[3 * t + 0], i1 = faces[3 * t + 1], i2 = faces[3 * t + 2];
        float x0 = verts[3 * i0], y0 = verts[3 * i0 + 1], z0 = verts[3 * i0 + 2];
        float x1 = verts[3 * i1], y1 = verts[3 * i1 + 1], z1 = verts[3 * i1 + 2];
        float x2 = verts[3 * i2], y2 = verts[3 * i2 + 1], z2 = verts[3 * i2 + 2];
        float w = (x1 - x0) * (y2 - y0) - (y1 - y0) * (x2 - x0);
        if (w >= EPS_W) {
            // area2d(A, p, C) = a*px + b*py + c
            aAB = y0 - y1; bAB = x1 - x0; cAB = y1 * (x0 - x1) - x1 * (y0 - y1);
            aCB = y1 - y2; bCB = x2 - x1; cCB = y2 * (x1 - x2) - x2 * (y1 - y2);
            aCA = y2 - y0; bCA = x0 - x2; cCA = y0 * (x2 - x0) - x0 * (y2 - y0);
            float invw = 1.0f / w;
            float dz0 = z0 - z2, dz1 = z1 - z2;
            az = (aCB * dz0 + aCA * dz1) * invw;
            bz = (bCB * dz0 + bCA * dz1) * invw;
            cz = z2 + (cCB * dz0 + cCA * dz1) * invw;
            int j0 = uvfaces[3 * t + 0], j1 = uvfaces[3 * t + 1], j2 = uvfaces[3 * t + 2];
            float u0 = uv[2 * j0] * 2.f - 1.f, w0 = uv[2 * j0 + 1] * 2.f - 1.f;
            float u1 = uv[2 * j1] * 2.f - 1.f, w1 = uv[2 * j1 + 1] * 2.f - 1.f;
            float u2 = uv[2 * j2] * 2.f - 1.f, w2 = uv[2 * j2 + 1] * 2.f - 1.f;
            float du0 = u0 - u2, du1 = u1 - u2, dv0 = w0 - w2, dv1 = w1 - w2;
            au = (aCB * du0 + aCA * du1) * invw;
            bu = (bCB * du0 + bCA * du1) * invw;
            cu = u2 + (cCB * du0 + cCA * du1) * invw;
            av = (aCB * dv0 + aCA * dv1) * invw;
            bv = (bCB * dv0 + bCA * dv1) * invw;
            cv = w2 + (cCB * dv0 + cCA * dv1) * invw;
        }
    }

    int g = t >> 2, j = t & 3;
    float* base = acoef + g * 64;
    float a_[4] = {aAB, aCB, aCA, az};
    float b_[4] = {bAB, bCB, bCA, bz};
    float c_[4] = {cAB, cCB, cCA, cz};
#pragma unroll
    for (int f = 0; f < 4; ++f) {
        int m = 4 * j + f;
        base[m * 2 + 0]        = a_[f];
        base[m * 2 + 1]        = b_[f];
        base[(m + 16) * 2 + 0] = c_[f];
        base[(m + 16) * 2 + 1] = 0.0f;
    }
    float* ub = uvcoef + t * 8;
    ub[0] = au; ub[1] = bu; ub[2] = cu;
    ub[3] = av; ub[4] = bv; ub[5] = cv;
    ub[6] = 0.f; ub[7] = 0.f;
}

// ---------------------------------------------------------------------------
// Kernel 3: fused raster + shade.  Each wave32 owns 16 pixels.
//  Prologue: wave 0 issues one TDM tensor_load_to_lds DMA staging the whole
//  coefficient array (ngroups x 256B) into LDS, waits TENSORcnt, block syncs.
//  Loop: 1 ds_load_b64 + 1 v_wmma_f32_16x16x4_f32 + per-lane lex-max (z,idx).
//  Half-waves (tris {0,1} vs {2,3} mod 4) merged once via shfl_xor(16).
// ---------------------------------------------------------------------------
__global__ __launch_bounds__(256) void raster_kernel(
    const float* __restrict__ acoef, const float* __restrict__ uvcoef,
    const float* __restrict__ zinit_p, const float* __restrict__ uvmap,
    float* __restrict__ out, int S, int T, int ngroups) {
    extern __shared__ float lcoef[];   // ngroups * 64 floats (256B per group)

    // --- TDM prologue: DMA acoef -> LDS (one DMA per block, wave 0 only) ---
    if (threadIdx.x < 32) {
        unsigned long long ga = (unsigned long long)(const void*)acoef;
        unsigned int lds_off = (unsigned int)(unsigned long long)(void*)lcoef;
        // D# group 0: count=1, lds_addr, global_addr[56:0], type=2
        u32x4 g0;
        g0.x = 1u;                        // count=1, is_restore=0, gather off
        g0.y = lds_off;                   // LDS byte address
        g0.z = (unsigned int)ga;          // global_addr[31:0]
        g0.w = (unsigned int)(ga >> 32) | 0x80000000u;  // [56:32] + type=2
        // D# group 1: data_size=8B; 2D tile: 32 elems (256B) x ngroups rows
        unsigned int td0 = 32u;               // tensor_dim0 (elements per row)
        unsigned int td1 = (unsigned int)ngroups;
        i32x8 g1;
        g1[0] = 0x00030000;                   // wg_mask=0, data_size=3 (8B)
        g1[1] = (int)((td0 & 0xffffu) << 16); // tensor_dim0[15:0] @ [31:16]
        g1[2] = (int)((td0 >> 16) | ((td1 & 0xffffu) << 16));
        g1[3] = (int)((td1 >> 16) | (32u << 16));       // tile_dim0=32 @ [31:16]
        g1[4] = (int)(td1 & 0xffffu);         // tile_dim1=ngroups, tile_dim2=0
        g1[5] = 32;                           // tensor_dim0_stride[31:0]=32
        g1[6] = 0;                            // stride hi / dim1_stride lo
        g1[7] = 0;
        i32x4 gz = {0, 0, 0, 0};              // groups 2/3 unused (2D tensor)
#if __clang_major__ >= 23
        i32x8 gz8 = {0, 0, 0, 0, 0, 0, 0, 0};
        __builtin_amdgcn_tensor_load_to_lds(g0, g1, gz, gz, gz8, 0);
#else
        __builtin_amdgcn_tensor_load_to_lds(g0, g1, gz, gz, 0);
#endif
        __builtin_amdgcn_s_wait_tensorcnt(0);
    }
    __syncthreads();

    const int lane = threadIdx.x & 31;
    const int waveGlobal = (blockIdx.x * blockDim.x + threadIdx.x) >> 5;
    const int n = lane & 15;
    const int npix = S * S;
    const int praw = waveGlobal * 16 + n;
    const int p = (praw < npix) ? praw : (npix - 1);
    const int pi = p / S, pj = p - pi * S;
    // pts after rot90: px = -1 + 2j/(S-1);  py = 1 - 2i/(S-1)
    const float inv = 2.0f / (float)(S - 1);
    const float px = -1.0f + (float)pj * inv;
    const float py = 1.0f - (float)pi * inv;

    // B matrix (4x16, column-major like A): column n = (px, py, 1, 0)
    v2f bmat;
    bmat.x = (lane < 16) ? px : 1.0f;   // K=0 / K=2
    bmat.y = (lane < 16) ? py : 0.0f;   // K=1 / K=3

    float bestz = *zinit_p;             // zbuf0 = min vertex z; ties pass (>=)
    int besti = -1;
    const int tbase = (lane < 16) ? 0 : 2;

    for (int g = 0; g < ngroups; ++g) {
        v2f amat = *(const v2f*)(lcoef + g * 64 + lane * 2);
        v8f c = {};
        // D[func][pixel]: rows = 4 tris x {pAB,pCB,pCA,z}, cols = 16 pixels
        v8f d = __builtin_amdgcn_wmma_f32_16x16x4_f32(
            false, amat, false, bmat, (short)0, c, false, false);
        int t0 = g * 4 + tbase;
        bool in0 = (d[0] > 0.f) && (d[1] > 0.f) && (d[2] > 0.f);
        float z0 = d[3];
        bool pass0 = in0 && (z0 >= bestz);
        bestz = pass0 ? z0 : bestz;
        besti = pass0 ? t0 : besti;
        bool in1 = (d[4] > 0.f) && (d[5] > 0.f) && (d[6] > 0.f);
        float z1 = d[7];
        bool pass1 = in1 && (z1 >= bestz);
        bestz = pass1 ? z1 : bestz;
        besti = pass1 ? (t0 + 1) : besti;
    }

    // Merge the two half-wave candidates for the same pixel: lex max (z, idx)
    float oz = __shfl_xor(bestz, 16, 32);
    int oi = __shfl_xor(besti, 16, 32);
    bool take = (oz > bestz) || ((oz == bestz) && (oi > besti));
    bestz = take ? oz : bestz;
    besti = take ? oi : besti;

    if (lane < 16 && praw < npix) {
        float r = 0.f, gch = 0.f, bch = 0.f, ach = 0.f;
        if (besti >= 0) {
            const float* uc = uvcoef + besti * 8;
            float u = uc[0] * px + uc[1] * py + uc[2];
            float v = uc[3] * px + uc[4] * py + uc[5];
            float ix = (u + 1.0f) * (float)(T - 1) * 0.5f;
            float iy = (v + 1.0f) * (float)(T - 1) * 0.5f;
            float ix0 = floorf(ix), iy0 = floorf(iy);
            float ix1 = ix0 + 1.0f, iy1 = iy0 + 1.0f;
            float wx1 = ix - ix0, wx0 = 1.0f - wx1;
            float wy1 = iy - iy0, wy0 = 1.0f - wy1;
            const float fT1 = (float)(T - 1);
            const int TT = T * T;
            auto tap = [&](float iyf, float ixf, float wgt) {
                bool inb = (ixf >= 0.f) && (ixf <= fT1) && (iyf >= 0.f) && (iyf <= fT1);
                int ii = (int)fminf(fmaxf(ixf, 0.f), fT1);
                int jj = (int)fminf(fmaxf(iyf, 0.f), fT1);
                float wm = inb ? wgt : 0.f;
                int off = jj * T + ii;
                r   += uvmap[off] * wm;
                gch += uvmap[TT + off] * wm;
                bch += uvmap[2 * TT + off] * wm;
            };
            tap(iy0, ix0, wy0 * wx0);
            tap(iy0, ix1, wy0 * wx1);
            tap(iy1, ix0, wy1 * wx0);
            tap(iy1, ix1, wy1 * wx1);
            ach = 1.0f;
        }
        const int SS = S * S;
        out[p] = r;
        out[SS + p] = gch;
        out[2 * SS + p] = bch;
        out[3 * SS + p] = ach;
    }
}

// ---------------------------------------------------------------------------
extern "C" void kernel_launch(void* const* d_in, const int* in_sizes, int n_in,
                              void* d_out, int out_size, void* d_ws, size_t ws_size,
                              hipStream_t stream) {
    const float* vertices = (const float*)d_in[0];
    const int* faces = (const int*)d_in[1];
    const float* uv = (const float*)d_in[2];
    const int* uvfaces = (const int*)d_in[3];
    const float* uvmap = (const float*)d_in[4];

    int nvert = in_sizes[0] / 3;
    int ntri = in_sizes[1] / 3;
    int T = (int)(sqrt((double)(in_sizes[4] / 3)) + 0.5);
    int S = (int)(sqrt((double)(out_size / 4)) + 0.5);

    int ntri_pad = (ntri + 3) & ~3;
    int ngroups = ntri_pad / 4;

    // Workspace: [0..4)=zinit, 256B-aligned A-coeff groups, then uv planes
    float* zinit = (float*)d_ws;
    float* acoef = (float*)((char*)d_ws + 256);
    float* uvcoef = acoef + (size_t)ngroups * 64;

    zinit_kernel<<<1, 256, 0, stream>>>(vertices, nvert, zinit);
    coef_kernel<<<(ntri_pad + 255) / 256, 256, 0, stream>>>(
        vertices, faces, uv, uvfaces, ntri, ntri_pad, acoef, uvcoef);

    int npix = S * S;
    int waves = (npix + 15) / 16;
    int blocks = (waves + 7) / 8;                 // 8 waves (256 thr) / block
    size_t shbytes = (size_t)ngroups * 256;       // staged coefficients in LDS
    raster_kernel<<<blocks, 256, shbytes, stream>>>(acoef, uvcoef, zinit, uvmap,
                                                    (float*)d_out, S, T, ngroups);
}